// VisDepthNet_75222057222807
// MI455X (gfx1250) — compile-verified
//
#include <hip/hip_runtime.h>
#include <math.h>

// Problem constants (from reference): B,V,C,D,H,W = 2,5,32,48,128,160
#define Bsz 2
#define NSRC 4
#define Cch 32
#define Dn 48
#define Hh 128
#define Ww 160
#define HW (Hh*Ww)
#define NPIX (Bsz*HW)          /* 40960  */
#define NSIM (Bsz*Dn*HW)       /* 1966080 */
#define EPSf 1e-5f

typedef float v2f __attribute__((ext_vector_type(2)));
typedef float v8f __attribute__((ext_vector_type(8)));

// Workspace layout (floats):
//   [0,192)   header: per (b,view) 16 floats: rot[9], trans[3], norv[3], pad
//             + fused BN coefs
//   simCur at 256, simSum at 256+NSIM, wsum at 256+2*NSIM
#define HDR_BNA0 128
#define HDR_BNB0 144
#define HDR_BNA1 160
#define HDR_BNB1 168
#define WS_SIMCUR 256
#define WS_SIMSUM (256 + NSIM)
#define WS_WSUM   (256 + 2*NSIM)

__device__ __forceinline__ void inv3(const float* A, float* o) {
  float a=A[0],b=A[1],c=A[2],d=A[3],e=A[4],f=A[5],g=A[6],h=A[7],i=A[8];
  float det = a*(e*i-f*h) - b*(d*i-f*g) + c*(d*h-e*g);
  float id = 1.f/det;
  o[0]= (e*i-f*h)*id; o[1]= -(b*i-c*h)*id; o[2]= (b*f-c*e)*id;
  o[3]= -(d*i-f*g)*id; o[4]= (a*i-c*g)*id; o[5]= -(a*f-c*d)*id;
  o[6]= (d*h-e*g)*id; o[7]= -(a*h-b*g)*id; o[8]= (a*e-b*d)*id;
}

// One small block. Computes per-(b,view) warp matrices, norv rows, fused BN.
__global__ void prep_kernel(const float* __restrict__ proj,
                            const float* __restrict__ g0p, const float* __restrict__ b0p,
                            const float* __restrict__ m0p, const float* __restrict__ v0p,
                            const float* __restrict__ g1p, const float* __restrict__ b1p,
                            const float* __restrict__ m1p, const float* __restrict__ v1p,
                            float* __restrict__ ws) {
  int t = threadIdx.x;
  if (t < Bsz*NSRC) {
    int b = t / NSRC, i = t % NSRC;
    const float* pmb = proj + (size_t)b*(5*2*16);   // (V,2,4,4) per batch
    const float* Eref = pmb;                        // view 0 extrinsic
    const float* Kref = pmb + 16;                   // view 0 intrinsic
    // fused ref (rows 0..2 of [K3@E3x4; 0001])
    float F[12];
    for (int r=0;r<3;r++) for (int c=0;c<4;c++) {
      float s=0.f; for (int j=0;j<3;j++) s += Kref[r*4+j]*Eref[j*4+c];
      F[r*4+c]=s;
    }
    // inverse of fused ref exploiting bottom row (0,0,0,1)
    float R3[9] = {F[0],F[1],F[2],F[4],F[5],F[6],F[8],F[9],F[10]};
    float t3[3] = {F[3],F[7],F[11]};
    float Ri[9]; inv3(R3,Ri);
    float it[3];
    for (int r=0;r<3;r++) it[r] = -(Ri[r*3]*t3[0]+Ri[r*3+1]*t3[1]+Ri[r*3+2]*t3[2]);
    // fused src
    const float* Es = pmb + (size_t)(i+1)*32;
    const float* Ks = Es + 16;
    float Gm[12];
    for (int r=0;r<3;r++) for (int c=0;c<4;c++) {
      float s=0.f; for (int j=0;j<3;j++) s += Ks[r*4+j]*Es[j*4+c];
      Gm[r*4+c]=s;
    }
    float* o = ws + (size_t)(b*NSRC+i)*16;
    for (int r=0;r<3;r++) {
      for (int c=0;c<3;c++)
        o[r*3+c] = Gm[r*4+0]*Ri[c] + Gm[r*4+1]*Ri[3+c] + Gm[r*4+2]*Ri[6+c];
      o[9+r] = Gm[r*4+0]*it[0] + Gm[r*4+1]*it[1] + Gm[r*4+2]*it[2] + Gm[r*4+3];
    }
    // norv = row 2 of (E_src[:3,:3] @ inv(E_ref[:3,:3]))
    float Er3[9] = {Eref[0],Eref[1],Eref[2],Eref[4],Eref[5],Eref[6],Eref[8],Eref[9],Eref[10]};
    float ERi[9]; inv3(Er3,ERi);
    for (int k=0;k<3;k++)
      o[12+k] = Es[8]*ERi[k] + Es[9]*ERi[3+k] + Es[10]*ERi[6+k];
  }
  if (t < 16) {
    float al = g0p[t]*rsqrtf(v0p[t]+EPSf);
    ws[HDR_BNA0+t] = al;
    ws[HDR_BNB0+t] = b0p[t] - m0p[t]*al;
  }
  if (t < 8) {
    float al = g1p[t]*rsqrtf(v1p[t]+EPSf);
    ws[HDR_BNA1+t] = al;
    ws[HDR_BNB1+t] = b1p[t] - m1p[t]*al;
  }
}

// Homography warp + bilinear gather + 32-channel dot with ref feature.
// One thread per (b,d,pixel). L2-resident gathers (2.6MB per plane << 192MB L2).
__global__ void __launch_bounds__(256)
warp_sim_kernel(const float* __restrict__ ref, const float* __restrict__ src,
                const float* __restrict__ depths, const float* __restrict__ ws,
                float* __restrict__ simCur, int view) {
  int tid = blockIdx.x*blockDim.x + threadIdx.x;
  if (tid >= NSIM) return;
  int n = tid % HW;
  int b = tid / (HW*Dn);
  int x = n % Ww, y = n / Ww;
  const float* hd = ws + (size_t)(b*NSRC+view)*16;
  float fx=(float)x, fy=(float)y;
  float rx = hd[0]*fx + hd[1]*fy + hd[2];
  float ry = hd[3]*fx + hd[4]*fy + hd[5];
  float rz = hd[6]*fx + hd[7]*fy + hd[8];
  float dep = depths[tid];                 // (B,D,H,W) flat == tid order
  float pz = rz*dep + hd[11];
  float gx = (rx*dep + hd[9])  / pz;
  float gy = (ry*dep + hd[10]) / pz;
  float x0 = floorf(gx), y0 = floorf(gy);
  float wx1 = gx-x0, wx0 = 1.f-wx1, wy1 = gy-y0, wy0 = 1.f-wy1;
  int off[4]; float msk[4];
  #pragma unroll
  for (int k=0;k<4;k++) {
    float xi = x0 + (float)(k&1), yi = y0 + (float)(k>>1);
    bool v = (xi>=0.f)&&(xi<=(float)(Ww-1))&&(yi>=0.f)&&(yi<=(float)(Hh-1));
    int xc = (int)fminf(fmaxf(xi,0.f),(float)(Ww-1));
    int yc = (int)fminf(fmaxf(yi,0.f),(float)(Hh-1));
    off[k] = yc*Ww+xc; msk[k] = v?1.f:0.f;
  }
  float w00=wx0*wy0*msk[0], w10=wx1*wy0*msk[1], w01=wx0*wy1*msk[2], w11=wx1*wy1*msk[3];
  const float* sb = src + ((size_t)view*Bsz + b)*(Cch*HW);
  const float* rb = ref + (size_t)b*(Cch*HW) + n;
  float acc = 0.f;
  #pragma unroll 4
  for (int c=0;c<Cch;c++) {
    const float* sc = sb + c*HW;
    float wv = w00*sc[off[0]] + w10*sc[off[1]] + w01*sc[off[2]] + w11*sc[off[3]];
    acc += rb[c*HW] * wv;
  }
  simCur[tid] = acc * (1.f/32.f);
}

// Pixelwise MLP (2->16->8->1, BN+ReLU, sigmoid, max over D) via WMMA f32 16x16x4.
// Mapping: M = output channels, N = 16 pixels per wave, K = input channels (chunks of 4).
// Layout assumptions (CDNA5 ISA 7.12.2):
//   A 16x4 f32: lane = M + 16*(K>>1), vgpr = K&1   (documented)
//   B 4x16 f32: lane = N + 16*(K>>1), vgpr = K&1   (mirrors A; assumed)
//   C/D 16x16:  vgpr r, lane l -> M = r + 8*(l>=16), N = l&15 (documented)
// D->B between layers is a half-wave lane shift: done with __shfl, no LDS.
__global__ void __launch_bounds__(128)
mlp_vw_kernel(const float* __restrict__ simCur, const float* __restrict__ normal,
              const float* __restrict__ w0p, const float* __restrict__ w1p,
              const float* __restrict__ w2p, const float* __restrict__ b2p,
              const float* __restrict__ ws, float* __restrict__ vwOut, int view) {
  const int lane = threadIdx.x & 31;
  const int m  = lane & 15;          // M row for A / N col for B,D
  const int hi = (lane >> 4) & 1;    // wave half
  const int wid = threadIdx.x >> 5;
  const int pix = (blockIdx.x*4 + wid)*16 + m;   // exact cover: 640 blocks * 64 px
  const int b = pix / HW, n = pix % HW;

  const float* hd = ws + (size_t)(b*NSRC+view)*16;
  float vis = hd[12]*normal[(b*3+0)*HW+n]
            + hd[13]*normal[(b*3+1)*HW+n]
            + hd[14]*normal[(b*3+2)*HW+n];

  // Hoisted fused-BN coefs (per lane, per D-row vgpr)
  float al0[8], be0[8], al1[8], be1[8];
  #pragma unroll
  for (int r=0;r<8;r++) {
    int mr = r + 8*hi;
    al0[r] = ws[HDR_BNA0+mr]; be0[r] = ws[HDR_BNB0+mr];
    al1[r] = hi ? 0.f : ws[HDR_BNA1+r];
    be1[r] = hi ? 0.f : ws[HDR_BNB1+r];
  }
  // Weights in WMMA A layout
  v2f A0; A0.x = hi?0.f:w0p[m*2+0]; A0.y = hi?0.f:w0p[m*2+1];        // K=2, pad to 4
  v2f A1[4], A2[2];
  #pragma unroll
  for (int kc=0;kc<4;kc++) {                                          // w1: 8x16, pad M
    A1[kc].x = (m<8)? w1p[m*16 + 4*kc + 2*hi]     : 0.f;
    A1[kc].y = (m<8)? w1p[m*16 + 4*kc + 1 + 2*hi] : 0.f;
  }
  #pragma unroll
  for (int kc=0;kc<2;kc++) {                                          // w2: 1x8
    A2[kc].x = (m==0)? w2p[4*kc + 2*hi]     : 0.f;
    A2[kc].y = (m==0)? w2p[4*kc + 1 + 2*hi] : 0.f;
  }
  const float bias2 = b2p[0];
  const float* simB = simCur + (size_t)b*Dn*HW + n;

  float pmax = -1.f;
  for (int d=0; d<Dn; d++) {
    float sim = simB[(size_t)d*HW];
    // Layer 0: h0(16x16) = w0(16x2pad4) @ x(2pad4 x 16)
    v2f B0; B0.x = hi?0.f:sim; B0.y = hi?0.f:vis;
    v8f h0 = {};
    h0 = __builtin_amdgcn_wmma_f32_16x16x4_f32(false, A0, false, B0, (short)0, h0, false, false);
    #pragma unroll
    for (int r=0;r<8;r++) h0[r] = fmaxf(0.f, h0[r]*al0[r] + be0[r]);  // BN+ReLU
    // Layer 1: acc1(8x16) = w1(8x16) @ h0(16x16), K=16 -> 4 WMMAs
    v8f acc1 = {};
    #pragma unroll
    for (int kc=0;kc<4;kc++) {
      float b0v, b1v;
      if (kc < 2) {   // rows 4kc..4kc+3 all <8: live in lanes 0-15 of vgpr=row
        float tC = __shfl(h0[4*kc+2], m, 32);
        float tD = __shfl(h0[4*kc+3], m, 32);
        b0v = hi ? tC : h0[4*kc+0];
        b1v = hi ? tD : h0[4*kc+1];
      } else {        // rows >=8: live in lanes 16-31 of vgpr=row-8
        float tA = __shfl(h0[4*kc+0-8], m+16, 32);
        float tB = __shfl(h0[4*kc+1-8], m+16, 32);
        b0v = hi ? h0[4*kc+2-8] : tA;
        b1v = hi ? h0[4*kc+3-8] : tB;
      }
      v2f Bv; Bv.x=b0v; Bv.y=b1v;
      acc1 = __builtin_amdgcn_wmma_f32_16x16x4_f32(false, A1[kc], false, Bv, (short)0, acc1, false, false);
    }
    float h1[8];
    #pragma unroll
    for (int r=0;r<8;r++) h1[r] = hi ? 0.f : fmaxf(0.f, acc1[r]*al1[r] + be1[r]);
    // Layer 2: acc2(1x16) = w2(1x8) @ h1(8x16), K=8 -> 2 WMMAs
    v8f acc2 = {};
    #pragma unroll
    for (int kc=0;kc<2;kc++) {
      float tC = __shfl(h1[4*kc+2], m, 32);
      float tD = __shfl(h1[4*kc+3], m, 32);
      v2f Bv; Bv.x = hi ? tC : h1[4*kc+0]; Bv.y = hi ? tD : h1[4*kc+1];
      acc2 = __builtin_amdgcn_wmma_f32_16x16x4_f32(false, A2[kc], false, Bv, (short)0, acc2, false, false);
    }
    float p = 1.f / (1.f + __expf(-(acc2[0] + bias2)));   // M=0 row: vgpr0, lanes 0-15
    pmax = fmaxf(pmax, p);
  }
  if (!hi) vwOut[(size_t)(b*NSRC + view)*HW + n] = pmax;
}

// sim_sum += sim*vw ; wsum += vw  (first view initializes -> no separate zero pass)
__global__ void __launch_bounds__(256)
accum_kernel(const float* __restrict__ simCur, const float* __restrict__ vwOut,
             float* __restrict__ simSum, float* __restrict__ wsum, int view) {
  int tid = blockIdx.x*blockDim.x + threadIdx.x;
  if (tid >= NPIX) return;
  int b = tid / HW, n = tid % HW;
  float vw = vwOut[(size_t)(b*NSRC+view)*HW + n];
  if (view == 0) wsum[tid] = 1e-5f + vw; else wsum[tid] += vw;
  size_t base = (size_t)b*Dn*HW + n;
  for (int d=0; d<Dn; d++) {
    size_t idx = base + (size_t)d*HW;
    float v = simCur[idx]*vw;
    simSum[idx] = (view==0) ? v : (simSum[idx]+v);
  }
}

// prob_volume_pre, softmax over D (online), argmax -> depth, max prob -> confidence,
// prob_volume, depth_values passthrough.
__global__ void __launch_bounds__(256)
final_kernel(const float* __restrict__ simSum, const float* __restrict__ wsum,
             const float* __restrict__ depths, const float* __restrict__ cwp,
             const float* __restrict__ cbp, float* __restrict__ out) {
  int tid = blockIdx.x*blockDim.x + threadIdx.x;
  if (tid >= NPIX) return;
  int b = tid / HW, n = tid % HW;
  float w  = wsum[tid];
  float cw = cwp[0], cb = cbp[0];
  float* outD   = out;                          // depth (B,H,W)
  float* outC   = out + NPIX;                   // confidence (B,H,W)
  float* outPV  = out + 2*(size_t)NPIX;         // prob_volume (B,D,H,W)
  float* outPVP = outPV + (size_t)NSIM;         // prob_volume_pre
  float* outDV  = outPVP + (size_t)NSIM;        // depth_values passthrough
  size_t base = (size_t)b*Dn*HW + n;
  float mmax = -__builtin_inff(), ssum = 0.f, bestdv = 0.f;
  for (int d=0; d<Dn; d++) {
    size_t idx = base + (size_t)d*HW;
    float pvp = simSum[idx]/w*cw + cb;
    float dv  = depths[idx];
    outPVP[idx] = pvp;
    outDV[idx]  = dv;
    if (pvp > mmax) { ssum = ssum*__expf(mmax-pvp) + 1.f; mmax = pvp; bestdv = dv; }
    else            { ssum += __expf(pvp - mmax); }
  }
  outD[tid] = bestdv;
  outC[tid] = 1.f/ssum;
  float inv = 1.f/ssum;
  for (int d=0; d<Dn; d++) {
    size_t idx = base + (size_t)d*HW;
    float pvp = simSum[idx]/w*cw + cb;
    outPV[idx] = __expf(pvp - mmax)*inv;
  }
}

extern "C" void kernel_launch(void* const* d_in, const int* in_sizes, int n_in,
                              void* d_out, int out_size, void* d_ws, size_t ws_size,
                              hipStream_t stream) {
  const float* ref    = (const float*)d_in[0];
  const float* src    = (const float*)d_in[1];
  const float* proj   = (const float*)d_in[2];
  const float* depths = (const float*)d_in[3];
  const float* normal = (const float*)d_in[4];
  const float* w0p = (const float*)d_in[5];
  const float* g0p = (const float*)d_in[6];
  const float* b0p = (const float*)d_in[7];
  const float* m0p = (const float*)d_in[8];
  const float* v0p = (const float*)d_in[9];
  const float* w1p = (const float*)d_in[10];
  const float* g1p = (const float*)d_in[11];
  const float* b1p = (const float*)d_in[12];
  const float* m1p = (const float*)d_in[13];
  const float* v1p = (const float*)d_in[14];
  const float* w2p = (const float*)d_in[15];
  const float* b2p = (const float*)d_in[16];
  const float* cwp = (const float*)d_in[17];
  const float* cbp = (const float*)d_in[18];

  float* out = (float*)d_out;
  float* wsf = (float*)d_ws;
  float* simCur = wsf + WS_SIMCUR;
  float* simSum = wsf + WS_SIMSUM;
  float* wsumP  = wsf + WS_WSUM;
  float* vwOut  = out + 2*(size_t)NPIX + 3*(size_t)NSIM;  // view_weights region of d_out

  prep_kernel<<<1, 64, 0, stream>>>(proj, g0p, b0p, m0p, v0p, g1p, b1p, m1p, v1p, wsf);

  for (int view = 0; view < NSRC; view++) {
    warp_sim_kernel<<<(NSIM+255)/256, 256, 0, stream>>>(ref, src, depths, wsf, simCur, view);
    mlp_vw_kernel<<<NPIX/64, 128, 0, stream>>>(simCur, normal, w0p, w1p, w2p, b2p, wsf, vwOut, view);
    accum_kernel<<<(NPIX+255)/256, 256, 0, stream>>>(simCur, vwOut, simSum, wsumP, view);
  }
  final_kernel<<<(NPIX+255)/256, 256, 0, stream>>>(simSum, wsumP, depths, cwp, cbp, out);
}